// Attention_13417477833308
// MI455X (gfx1250) — compile-verified
//
#include <hip/hip_runtime.h>
#include <math.h>

#define B_    32
#define NH    32
#define HD    128
#define NKV   8
#define GQ    4            // NH / NKV
#define BS    256
#define MB    16
#define MAXTOK (MB * BS)   // 4096
#define SCALE_F 0.08838834764831845f  // 1/sqrt(128)

typedef __attribute__((ext_vector_type(2))) float v2f;
typedef __attribute__((ext_vector_type(8))) float v8f;

__global__ __launch_bounds__(256) void paged_attn_wmma_f32(
    const float* __restrict__ q,
    const float* __restrict__ knew,
    const float* __restrict__ vnew,
    const float* __restrict__ kc,
    const float* __restrict__ vc,
    const int* __restrict__ slot_mapping,
    const int* __restrict__ block_tables,
    const int* __restrict__ context_lens,
    float* __restrict__ out)
{
    // LDS: overlay map (token -> batch owning a fresh k/v row), block table,
    // per-wave P transpose scratch (row stride 20 dwords to dodge bank conflicts),
    // and cross-wave softmax-combine buffers.
    __shared__ short ovr[MAXTOK];
    __shared__ int   sbt[MB];
    __shared__ float pxp[8][16 * 20];
    __shared__ float smO[8][GQ][HD];
    __shared__ float smM[8][GQ];
    __shared__ float smL[8][GQ];

    const int kv   = blockIdx.x;   // 0..7
    const int b    = blockIdx.y;   // 0..31
    const int tid  = threadIdx.x;  // 0..255
    const int w    = tid >> 5;     // wave id 0..7 (wave32)
    const int lane = tid & 31;
    const int h    = lane >> 4;    // lane half
    const int c    = lane & 15;    // position within half

    // ---------- build overlay table: which gathered tokens were just written ----------
    for (int i = tid; i < MAXTOK; i += 256) ovr[i] = -1;
    if (tid < MB) sbt[tid] = block_tables[b * MB + tid];
    __syncthreads();
    for (int p = tid; p < B_ * MB; p += 256) {
        int b2 = p >> 4, i = p & (MB - 1);
        int slot = slot_mapping[b2];
        if (sbt[i] == (slot >> 8)) ovr[i * BS + (slot & (BS - 1))] = (short)b2;
    }
    __syncthreads();

    const int L = context_lens[b];
    const int T = (L + 15) >> 4;   // number of 16-token tiles

    // ---------- Q A-fragments (heads on M, 4 valid rows), SCALE folded in ----------
    v2f qa[32];
    {
        const bool hv = (c < GQ);
        const float* qb = q + ((size_t)b * NH + kv * GQ + (hv ? c : 0)) * HD;
        #pragma unroll
        for (int s = 0; s < 32; ++s) {
            if (hv) {
                v2f t = *(const v2f*)(qb + 4 * s + 2 * h);
                qa[s].x = t.x * SCALE_F;
                qa[s].y = t.y * SCALE_F;
            } else {
                qa[s].x = 0.f; qa[s].y = 0.f;
            }
        }
    }

    v8f acc[8];
    #pragma unroll
    for (int j = 0; j < 8; ++j) acc[j] = {};
    float m_r[4] = {-1e30f, -1e30f, -1e30f, -1e30f};
    float l_r[4] = {0.f, 0.f, 0.f, 0.f};

    for (int tile = w; tile < T; tile += 8) {
        const int tb   = tile << 4;
        const int phys = sbt[tb >> 8];           // 16 tokens never straddle a block
        const int boff = tb & (BS - 1);

        // ---- K B-fragments: lane covers token tb+c, dims pair per step ----
        const int tokK = tb + c;
        const int obK  = ovr[tokK];
        const float* kb_base = (obK >= 0)
            ? (knew + ((size_t)obK * NKV + kv) * HD)
            : (kc + (((size_t)phys * BS + boff + c) * NKV + kv) * HD);
        v2f kb[32];
        #pragma unroll
        for (int s = 0; s < 32; ++s)
            kb[s] = *(const v2f*)(kb_base + 4 * s + 2 * h);

        // ---- S = Q * K^T  (16 heads x 16 tokens), 32x f32 WMMA, K=4 each ----
        v8f S = {};
        #pragma unroll
        for (int s = 0; s < 32; ++s)
            S = __builtin_amdgcn_wmma_f32_16x16x4_f32(false, qa[s], false, kb[s],
                                                      (short)0, S, false, false);

        // ---- mask invalid token columns, online softmax on rows 0..3 ----
        const bool tvalid = (tokK < L);
        #pragma unroll
        for (int r = 0; r < 8; ++r) S[r] = tvalid ? S[r] : -1e30f;

        float p8[8];
        float alpha[4];
        #pragma unroll
        for (int r = 0; r < 4; ++r) {
            float mx = S[r];
            mx = fmaxf(mx, __shfl_xor(mx, 1, 32));
            mx = fmaxf(mx, __shfl_xor(mx, 2, 32));
            mx = fmaxf(mx, __shfl_xor(mx, 4, 32));
            mx = fmaxf(mx, __shfl_xor(mx, 8, 32));
            float mnew = fmaxf(m_r[r], mx);
            alpha[r] = __expf(m_r[r] - mnew);
            float pe = __expf(S[r] - mnew);
            float rs = pe;
            rs += __shfl_xor(rs, 1, 32);
            rs += __shfl_xor(rs, 2, 32);
            rs += __shfl_xor(rs, 4, 32);
            rs += __shfl_xor(rs, 8, 32);
            l_r[r] = l_r[r] * alpha[r] + rs;
            m_r[r] = mnew;
            p8[r] = pe;
        }
        #pragma unroll
        for (int r = 4; r < 8; ++r) p8[r] = 0.f;

        // rescale running output accumulators
        #pragma unroll
        for (int j = 0; j < 8; ++j) {
            #pragma unroll
            for (int r = 0; r < 8; ++r) acc[j][r] *= alpha[r & 3];
        }

        // ---- transpose P (C-layout -> A-layout) through per-wave LDS ----
        float* pb = &pxp[w][0];
        #pragma unroll
        for (int r = 0; r < 8; ++r)
            pb[(r + 8 * h) * 20 + c] = p8[r];
        asm volatile("s_wait_dscnt 0" ::: "memory");
        v2f pa[4];
        #pragma unroll
        for (int s = 0; s < 4; ++s)
            pa[s] = *(const v2f*)(pb + c * 20 + 4 * s + 2 * h);
        asm volatile("" ::: "memory");

        // ---- V B-fragments: token rows per (s,v,half), dim col = 16j + c ----
        float vf[4][2][8];
        #pragma unroll
        for (int s = 0; s < 4; ++s) {
            #pragma unroll
            for (int v = 0; v < 2; ++v) {
                const int to   = 4 * s + v + 2 * h;
                const int obV  = ovr[tb + to];
                const float* vrow = (obV >= 0)
                    ? (vnew + ((size_t)obV * NKV + kv) * HD)
                    : (vc + (((size_t)phys * BS + boff + to) * NKV + kv) * HD);
                #pragma unroll
                for (int j = 0; j < 8; ++j) vf[s][v][j] = vrow[16 * j + c];
            }
        }

        // ---- O += P * V : 8 output slices x 4 K-steps of f32 WMMA ----
        #pragma unroll
        for (int j = 0; j < 8; ++j) {
            #pragma unroll
            for (int s = 0; s < 4; ++s) {
                v2f bv; bv.x = vf[s][0][j]; bv.y = vf[s][1][j];
                acc[j] = __builtin_amdgcn_wmma_f32_16x16x4_f32(false, pa[s], false, bv,
                                                               (short)0, acc[j], false, false);
            }
        }
    }

    // ---------- per-wave partials to LDS ----------
    if (lane == 0) {
        #pragma unroll
        for (int r = 0; r < GQ; ++r) { smM[w][r] = m_r[r]; smL[w][r] = l_r[r]; }
    }
    if (h == 0) {
        #pragma unroll
        for (int r = 0; r < GQ; ++r) {
            #pragma unroll
            for (int j = 0; j < 8; ++j) smO[w][r][16 * j + c] = acc[j][r];
        }
    }
    __syncthreads();

    // ---------- cross-wave softmax merge + store ----------
    for (int idx = tid; idx < GQ * HD; idx += 256) {
        const int g = idx >> 7;
        const int d = idx & (HD - 1);
        float gm = -1e30f;
        #pragma unroll
        for (int w2 = 0; w2 < 8; ++w2) gm = fmaxf(gm, smM[w2][g]);
        float Lt = 0.f, ac = 0.f;
        #pragma unroll
        for (int w2 = 0; w2 < 8; ++w2) {
            float sc = __expf(smM[w2][g] - gm);
            Lt += smL[w2][g] * sc;
            ac += smO[w2][g][d] * sc;
        }
        out[((size_t)b * NH + kv * GQ + g) * HD + d] = ac / Lt;
    }
}

extern "C" void kernel_launch(void* const* d_in, const int* in_sizes, int n_in,
                              void* d_out, int out_size, void* d_ws, size_t ws_size,
                              hipStream_t stream) {
    const float* q  = (const float*)d_in[0];
    const float* k  = (const float*)d_in[1];
    const float* v  = (const float*)d_in[2];
    const float* kc = (const float*)d_in[3];
    const float* vc = (const float*)d_in[4];
    const int* slot = (const int*)d_in[5];
    const int* bt   = (const int*)d_in[6];
    const int* cl   = (const int*)d_in[7];

    dim3 grid(NKV, B_);   // 8 x 32 = 256 workgroups, 8 waves each
    paged_attn_wmma_f32<<<grid, 256, 0, stream>>>(q, k, v, kc, vc, slot, bt, cl,
                                                  (float*)d_out);
}